// KnowledgeMLP_v1_76441827934654
// MI455X (gfx1250) — compile-verified
//
#include <hip/hip_runtime.h>

typedef __attribute__((ext_vector_type(2))) float v2f;
typedef __attribute__((ext_vector_type(8))) float v8f;

// D = A(16x4) * B(4x16) + C, fp32 WMMA (CDNA5 V_WMMA_F32_16X16X4_F32)
__device__ __forceinline__ v8f wmma_f32(v2f a, v2f b, v8f c) {
    return __builtin_amdgcn_wmma_f32_16x16x4_f32(
        /*neg_a=*/false, a, /*neg_b=*/false, b,
        /*c_mod=*/(short)0, c, /*reuse_a=*/false, /*reuse_b=*/false);
}

// ---------------------------------------------------------------------------
// Generic C[M,N] = act(A[M,K] @ W[K,N] + bias).  Row-major everything.
// Block = 128 threads = 4 waves in a 2x2 arrangement -> 64x64 block tile,
// each wave owns a 32x32 tile (4 accumulators: 8B of operands per WMMA/lane).
// Requires M%64==0, N%64==0, K%4==0 (true for all calls here).
// ---------------------------------------------------------------------------
__global__ __launch_bounds__(128) void gemm_bias_act(
    const float* __restrict__ A, const float* __restrict__ W,
    const float* __restrict__ bias, float* __restrict__ C,
    int M, int K, int N, int relu)
{
    const int lane = threadIdx.x & 31;
    const int wid  = threadIdx.x >> 5;
    const int ln16 = lane & 15;
    const int half = lane >> 4;                 // 0: lanes 0-15, 1: lanes 16-31
    const int m0   = blockIdx.x * 64 + (wid >> 1) * 32;
    const int n0   = blockIdx.y * 64 + (wid & 1) * 32;

    v8f c00 = {}, c01 = {}, c10 = {}, c11 = {};

    const float* arow0 = A + (size_t)(m0 + ln16) * K;        // M-rows 0..15
    const float* arow1 = A + (size_t)(m0 + 16 + ln16) * K;   // M-rows 16..31

    for (int kk = 0; kk < K; kk += 4) {
        const int ka = kk + 2 * half;           // this lane-half's K columns
        v2f a0 = { arow0[ka], arow0[ka + 1] };
        v2f a1 = { arow1[ka], arow1[ka + 1] };
        const float* wr = W + (size_t)ka * N + n0 + ln16;
        v2f b0 = { wr[0],  wr[N] };             // N cols  0..15, K rows ka, ka+1
        v2f b1 = { wr[16], wr[N + 16] };        // N cols 16..31
        c00 = wmma_f32(a0, b0, c00);
        c01 = wmma_f32(a0, b1, c01);
        c10 = wmma_f32(a1, b0, c10);
        c11 = wmma_f32(a1, b1, c11);
    }

    const float bn0 = bias[n0 + ln16];
    const float bn1 = bias[n0 + 16 + ln16];
#pragma unroll
    for (int v = 0; v < 8; ++v) {
        const int mA = m0 + v + 8 * half;       // rows for c0x
        const int mB = mA + 16;                 // rows for c1x
        float e00 = c00[v] + bn0, e01 = c01[v] + bn1;
        float e10 = c10[v] + bn0, e11 = c11[v] + bn1;
        if (relu) {
            e00 = fmaxf(e00, 0.f); e01 = fmaxf(e01, 0.f);
            e10 = fmaxf(e10, 0.f); e11 = fmaxf(e11, 0.f);
        }
        C[(size_t)mA * N + n0 + ln16]      = e00;
        C[(size_t)mA * N + n0 + 16 + ln16] = e01;
        C[(size_t)mB * N + n0 + ln16]      = e10;
        C[(size_t)mB * N + n0 + 16 + ln16] = e11;
    }
}

// ---------------------------------------------------------------------------
// Knowledge branch: for each batch row b,
//   S[b,:] = H4[b,:] + sum_k mask[b,k] * ReLU(kg[idx[b,k],:] @ Wkg + bkg)
// One wave per b; Wkg (64x64) staged in LDS once per block.  The gathered
// rows form the WMMA M dimension (4 M-tiles of 16, rows 50..63 masked to 0).
// Mask is applied AFTER ReLU (ReLU(bias) != 0 in general), then reduced over
// the M dimension in-register plus one shfl_xor(16) across lane halves.
// ---------------------------------------------------------------------------
__global__ __launch_bounds__(128) void knowledge_kernel(
    const float* __restrict__ kg, const int* __restrict__ idx,
    const int* __restrict__ mask, const float* __restrict__ Wkg,
    const float* __restrict__ bkg, const float* __restrict__ H4,
    float* __restrict__ S)
{
    __shared__ float lds_w[64 * 64];
    __shared__ float lds_b[64];
    __shared__ float lds_mask[4][64];

    for (int i = threadIdx.x; i < 64 * 64; i += 128) lds_w[i] = Wkg[i];
    if (threadIdx.x < 64) lds_b[threadIdx.x] = bkg[threadIdx.x];

    const int lane = threadIdx.x & 31;
    const int wid  = threadIdx.x >> 5;
    const int ln16 = lane & 15;
    const int half = lane >> 4;
    const int b    = blockIdx.x * 4 + wid;      // batch row owned by this wave

    for (int m = lane; m < 64; m += 32)
        lds_mask[wid][m] = (m < 50) ? (float)mask[b * 50 + m] : 0.f;
    __syncthreads();

    v8f run0 = {}, run1 = {}, run2 = {}, run3 = {};

    for (int mt = 0; mt < 4; ++mt) {
        const int m   = mt * 16 + ln16;
        const int row = (m < 50) ? idx[b * 50 + m] : 0;   // clamp padded rows
        const float* arow = kg + (size_t)row * 64;

        v8f t0 = {}, t1 = {}, t2 = {}, t3 = {};
        for (int kk = 0; kk < 64; kk += 4) {
            const int ka = kk + 2 * half;
            v2f a = { arow[ka], arow[ka + 1] };
            const float* wr = &lds_w[ka * 64 + ln16];
            v2f b0 = { wr[0],  wr[64] };
            v2f b1 = { wr[16], wr[80] };
            v2f b2 = { wr[32], wr[96] };
            v2f b3 = { wr[48], wr[112] };
            t0 = wmma_f32(a, b0, t0);
            t1 = wmma_f32(a, b1, t1);
            t2 = wmma_f32(a, b2, t2);
            t3 = wmma_f32(a, b3, t3);
        }
#pragma unroll
        for (int v = 0; v < 8; ++v) {
            const float mk = lds_mask[wid][mt * 16 + v + 8 * half];
            run0[v] += fmaxf(t0[v] + lds_b[ 0 + ln16], 0.f) * mk;
            run1[v] += fmaxf(t1[v] + lds_b[16 + ln16], 0.f) * mk;
            run2[v] += fmaxf(t2[v] + lds_b[32 + ln16], 0.f) * mk;
            run3[v] += fmaxf(t3[v] + lds_b[48 + ln16], 0.f) * mk;
        }
    }

    float s0 = 0.f, s1 = 0.f, s2 = 0.f, s3 = 0.f;
#pragma unroll
    for (int v = 0; v < 8; ++v) { s0 += run0[v]; s1 += run1[v]; s2 += run2[v]; s3 += run3[v]; }
    s0 += __shfl_xor(s0, 16, 32);
    s1 += __shfl_xor(s1, 16, 32);
    s2 += __shfl_xor(s2, 16, 32);
    s3 += __shfl_xor(s3, 16, 32);

    if (half == 0) {
        const size_t base = (size_t)b * 64;
        S[base +  0 + ln16] = s0 + H4[base +  0 + ln16];
        S[base + 16 + ln16] = s1 + H4[base + 16 + ln16];
        S[base + 32 + ln16] = s2 + H4[base + 32 + ln16];
        S[base + 48 + ln16] = s3 + H4[base + 48 + ln16];
    }
}

// ---------------------------------------------------------------------------
// Launch: 4 layer1 GEMMs -> knowledge (fused +h) -> layer2 GEMM.
// Workspace ping-pong: bufA = 8192*128 f32, bufB = 8192*512 f32 (~21 MB).
// ---------------------------------------------------------------------------
extern "C" void kernel_launch(void* const* d_in, const int* in_sizes, int n_in,
                              void* d_out, int out_size, void* d_ws, size_t ws_size,
                              hipStream_t stream)
{
    (void)in_sizes; (void)n_in; (void)out_size; (void)ws_size;
    const float* x    = (const float*)d_in[0];
    const float* kg   = (const float*)d_in[1];
    const int*   idx  = (const int*)  d_in[2];
    const int*   mask = (const int*)  d_in[3];
    const float* Wkg  = (const float*)d_in[4];
    const float* bkg  = (const float*)d_in[5];
    const float* W1a  = (const float*)d_in[6];
    const float* b1a  = (const float*)d_in[7];
    const float* W1b  = (const float*)d_in[8];
    const float* b1b  = (const float*)d_in[9];
    const float* W1c  = (const float*)d_in[10];
    const float* b1c  = (const float*)d_in[11];
    const float* W1d  = (const float*)d_in[12];
    const float* b1d  = (const float*)d_in[13];
    const float* W2   = (const float*)d_in[14];
    const float* b2   = (const float*)d_in[15];
    float* out = (float*)d_out;

    float* bufA = (float*)d_ws;                     // 8192*128 floats
    float* bufB = bufA + (size_t)8192 * 128;        // 8192*512 floats

    const dim3 blk(128);
    // H1 = relu(x @ W1a + b1a)           [8192,512]x[512,128]
    gemm_bias_act<<<dim3(128,  2), blk, 0, stream>>>(x,    W1a, b1a, bufA, 8192, 512,  128, 1);
    // H2 = relu(H1 @ W1b + b1b)          [8192,128]x[128,512]
    gemm_bias_act<<<dim3(128,  8), blk, 0, stream>>>(bufA, W1b, b1b, bufB, 8192, 128,  512, 1);
    // H3 = relu(H2 @ W1c + b1c)          [8192,512]x[512,128]   (reuse bufA)
    gemm_bias_act<<<dim3(128,  2), blk, 0, stream>>>(bufB, W1c, b1c, bufA, 8192, 512,  128, 1);
    // H4 = relu(H3 @ W1d + b1d)          [8192,128]x[128,64]    (reuse bufB)
    gemm_bias_act<<<dim3(128,  1), blk, 0, stream>>>(bufA, W1d, b1d, bufB, 8192, 128,   64, 1);
    // S = knowledge + H4                 (reuse bufA)
    knowledge_kernel<<<dim3(2048), blk, 0, stream>>>(kg, idx, mask, Wkg, bkg, bufB, bufA);
    // out = S @ W2 + b2                  [8192,64]x[64,1024]
    gemm_bias_act<<<dim3(128, 16), blk, 0, stream>>>(bufA, W2,  b2,  out,  8192,  64, 1024, 0);
}